// SATNetLayer_87428354277597
// MI455X (gfx1250) — compile-verified
//
#include <hip/hip_runtime.h>
#include <math.h>

typedef __attribute__((ext_vector_type(2))) float v2f;
typedef __attribute__((ext_vector_type(8))) float v8f;

#define BATCH  16
#define NVARS  512
#define MCLS   600
#define NTOT   813
#define NPAD   816            // NTOT rounded to 16
#define KDIM   41
#define NSWEEP 10
#define EPSF   1e-10f
#define PI_F   3.14159265358979323846f

// workspace offsets (in floats), all 16B aligned
#define OFF_VR    0           // 41*813 = 33333 -> pad 33344
#define OFF_P     33344       // 41*512 = 20992
#define OFF_GDIAG 54336       // 816
#define OFF_G     55152       // 816*816 = 665856 (padded Gram)
#define OFF_V     721008      // 16*41*813 = 533328
#define OFF_SPAD  OFF_V       // 600*816 = 489600, aliased: used only before V init

#define LDS_STRIDE 817        // odd stride: conflict-free LDS column walks

// ---- CDNA5 async global->LDS copy (ASYNCcnt) with safe fallback ----------
#if __has_builtin(__builtin_amdgcn_global_load_async_to_lds_b128)
#define HAVE_ASYNC_LDS 1
#else
#define HAVE_ASYNC_LDS 0
#endif

#if HAVE_ASYNC_LDS
typedef int v4i_vs __attribute__((vector_size(16)));
typedef v4i_vs __attribute__((address_space(1)))* gp128_t;   // global b128 ptr
typedef v4i_vs __attribute__((address_space(3)))* lp128_t;   // LDS b128 ptr
#endif

__device__ __forceinline__ void grow_fetch(const float* __restrict__ src,
                                           float* dst, int lane) {
#if HAVE_ASYNC_LDS
#pragma unroll
    for (int it = 0; it < 7; ++it) {
        int n = lane * 4 + it * 128;
        if (n < NPAD)
            __builtin_amdgcn_global_load_async_to_lds_b128(
                (gp128_t)(src + n), (lp128_t)(dst + n), 0, 0);
    }
#else
    for (int n = lane; n < NPAD; n += 32) dst[n] = src[n];
#endif
}

__device__ __forceinline__ void grow_wait() {
#if HAVE_ASYNC_LDS
#if __has_builtin(__builtin_amdgcn_s_wait_asynccnt)
    __builtin_amdgcn_s_wait_asynccnt(0);
#else
    asm volatile("s_wait_asynccnt 0x0" ::: "memory");
#endif
#endif
}

// ---------------------------------------------------------------- kernel 0
// Zero-padded copy of S: (600,813) -> (600,816)
__global__ void satnet_pad_s(const float* __restrict__ S,
                             float* __restrict__ Spad) {
    int idx = blockIdx.x * blockDim.x + threadIdx.x;
    if (idx >= MCLS * NPAD) return;
    int r = idx / NPAD, c = idx % NPAD;
    Spad[idx] = (c < NTOT) ? S[r * NTOT + c] : 0.f;
}

// ---------------------------------------------------------------- kernel 1
// Vr = v_rand / (||col|| + eps)   (column norms over k=41)
__global__ void satnet_norm_cols(const float* __restrict__ vr_in,
                                 float* __restrict__ Vr) {
    int j = blockIdx.x * blockDim.x + threadIdx.x;
    if (j >= NTOT) return;
    float ss = 0.f;
#pragma unroll
    for (int k = 0; k < KDIM; ++k) {
        float v = vr_in[k * NTOT + j];
        ss += v * v;
    }
    float inv = 1.f / (sqrtf(ss) + EPSF);
#pragma unroll
    for (int k = 0; k < KDIM; ++k)
        Vr[k * NTOT + j] = vr_in[k * NTOT + j] * inv;
}

// ---------------------------------------------------------------- kernel 2
// P_j = normalize(Vr_{1+j} - v0 (v0 . Vr_{1+j})), j = 0..511
__global__ void satnet_project(const float* __restrict__ Vr,
                               float* __restrict__ P) {
    int j = blockIdx.x * blockDim.x + threadIdx.x;
    if (j >= NVARS) return;
    int col = 1 + j;
    float p[KDIM];
    float d = 0.f;
#pragma unroll
    for (int k = 0; k < KDIM; ++k) {
        float vv  = Vr[k * NTOT + col];
        float v0k = Vr[k * NTOT];
        p[k] = vv;
        d += v0k * vv;
    }
    float ss = 0.f;
#pragma unroll
    for (int k = 0; k < KDIM; ++k) {
        float pk = p[k] - Vr[k * NTOT] * d;
        p[k] = pk;
        ss += pk * pk;
    }
    float inv = 1.f / (sqrtf(ss) + EPSF);
#pragma unroll
    for (int k = 0; k < KDIM; ++k)
        P[k * NVARS + j] = p[k] * inv;
}

// ---------------------------------------------------------------- kernel 3
// G = S^T S via V_WMMA_F32_16X16X4_F32.  One 16x16 tile per wave.
// Spad is zero-padded -> every load and store is unconditional.
#define GTILES (NPAD / 16)              // 51
__global__ __launch_bounds__(256)
void satnet_gram_wmma(const float* __restrict__ Spad,
                      float* __restrict__ G) {
    int tid  = threadIdx.x;
    int wave = tid >> 5;
    int lane = tid & 31;
    int tile = blockIdx.x * 8 + wave;
    if (tile >= GTILES * GTILES) return;        // wave-uniform branch
    int i0 = (tile / GTILES) * 16;
    int j0 = (tile % GTILES) * 16;

    int laneN = lane & 15;
    int rsel  = (lane >> 4) * 2;                // lanes 0-15: K=0,1  lanes 16-31: K=2,3
    int colA  = i0 + laneN;
    int colB  = j0 + laneN;

    v8f acc = {};
    for (int kk = 0; kk < MCLS; kk += 4) {
        const float* s0 = Spad + (size_t)(kk + rsel) * NPAD;
        v2f a, b;
        a.x = s0[colA];
        a.y = s0[NPAD + colA];
        b.x = s0[colB];
        b.y = s0[NPAD + colB];
        // D = A(16x4) * B(4x16) + C, full f32
        acc = __builtin_amdgcn_wmma_f32_16x16x4_f32(
            false, a, false, b, (short)0, acc, false, false);
    }

    float* gout = G + (size_t)(i0 + (lane >> 4) * 8) * NPAD + j0 + laneN;
#pragma unroll
    for (int r = 0; r < 8; ++r)
        gout[(size_t)r * NPAD] = acc[r];        // coalesced, unconditional
}

// ---------------------------------------------------------------- kernel 3b
// gdiag = diag(G) (branch-free extraction; avoids exec games in the GEMM)
__global__ void satnet_gdiag(const float* __restrict__ G,
                             float* __restrict__ gdiag) {
    int j = blockIdx.x * blockDim.x + threadIdx.x;
    if (j >= NPAD) return;
    gdiag[j] = G[(size_t)j * NPAD + j];
}

// ---------------------------------------------------------------- kernel 4
// Initialize V[b,:, :]:  col0 = v0, cols 1..512 = -c*v0 + s*P, rest = Vr aux
__global__ void satnet_init_v(const float* __restrict__ x,
                              const float* __restrict__ Vr,
                              const float* __restrict__ P,
                              float* __restrict__ V) {
    int idx = blockIdx.x * blockDim.x + threadIdx.x;
    if (idx >= BATCH * NTOT) return;
    int b = idx / NTOT;
    int j = idx % NTOT;
    float* Vb = V + (size_t)b * KDIM * NTOT;
    if (j == 0) {
#pragma unroll
        for (int k = 0; k < KDIM; ++k)
            Vb[(size_t)k * NTOT] = Vr[k * NTOT];
    } else if (j <= NVARS) {
        float z = x[b * NVARS + (j - 1)];
        float c = cosf(PI_F * z);
        float s = sinf(PI_F * z);
#pragma unroll
        for (int k = 0; k < KDIM; ++k)
            Vb[(size_t)k * NTOT + j] = -c * Vr[k * NTOT] + s * P[k * NVARS + (j - 1)];
    } else {
#pragma unroll
        for (int k = 0; k < KDIM; ++k)
            Vb[(size_t)k * NTOT + j] = Vr[k * NTOT + j];
    }
}

// ---------------------------------------------------------------- kernel 5
// Per-batch Gauss-Seidel sweeps with V resident in LDS (CDNA5: 320KB/WGP).
// G rows double-buffered in LDS via async global->LDS copies (ASYNCcnt).
// blockDim = 384 = 48 k-slots x 8 partial ranges.
#define SWT 384
#define SM_GROW 0                         // 2 x 816 floats (16B aligned)
#define SM_V0   (2 * NPAD)                // 48
#define SM_GD   (2 * NPAD + 48)           // 304 (gdiag for i = 513..812)
#define SM_RED  (2 * NPAD + 48 + 304)     // 384
#define SM_GBUF (2 * NPAD + 48 + 304 + 384)          // 48
#define SM_VL   (2 * NPAD + 48 + 304 + 384 + 48)     // 41*817
__global__ __launch_bounds__(SWT)
void satnet_sweep(const float* __restrict__ Vr,
                  const float* __restrict__ G,
                  const float* __restrict__ gdiag,
                  const float* __restrict__ V,
                  float* __restrict__ out) {
    extern __shared__ float sm[];
    float* grow = sm + SM_GROW;
    float* v0l  = sm + SM_V0;
    float* gdl  = sm + SM_GD;
    float* red  = sm + SM_RED;
    float* gbuf = sm + SM_GBUF;
    float* Vl   = sm + SM_VL;

    int b    = blockIdx.x;
    int tid  = threadIdx.x;
    int part = tid / 48;                 // 0..7
    int ks   = tid % 48;                 // 0..47 (41 live)
    int nb   = part * 102;
    int ne   = (nb + 102 < NTOT) ? nb + 102 : NTOT;

    // stage V[b] into LDS
    const float* Vb = V + (size_t)b * KDIM * NTOT;
    for (int idx = tid; idx < KDIM * NTOT; idx += SWT) {
        int k = idx / NTOT, j = idx % NTOT;
        Vl[k * LDS_STRIDE + j] = Vb[(size_t)k * NTOT + j];
    }
    if (tid < KDIM) v0l[tid] = Vr[tid * NTOT];
    if (tid < (NTOT - NVARS - 1)) gdl[tid] = gdiag[NVARS + 1 + tid];
    // prologue: async-fetch first free row (i = 513) into buffer 0
    if (tid < 32) {
        grow_fetch(G + (size_t)(NVARS + 1) * NPAD, grow, tid);
        grow_wait();
    }
    __syncthreads();

    int cur = 0;
    for (int sw = 0; sw < NSWEEP; ++sw) {
        for (int i = NVARS + 1; i < NTOT; ++i) {
            // kick off async copy of the next row into the other buffer
            int nxt = (i + 1 < NTOT) ? i + 1 : NVARS + 1;
            if (tid < 32)
                grow_fetch(G + (size_t)nxt * NPAD, grow + (cur ^ 1) * NPAD, tid);

            const float* Gi = grow + cur * NPAD;
            float partial = 0.f;
            if (ks < KDIM) {
                const float* vrow = Vl + ks * LDS_STRIDE;
                for (int n = nb; n < ne; ++n)
                    partial = fmaf(vrow[n], Gi[n], partial);
            }
            red[tid] = partial;
            __syncthreads();

            if (part == 0 && ks < KDIM) {
                float g = 0.f;
#pragma unroll
                for (int p = 0; p < 8; ++p) g += red[p * 48 + ks];
                g -= gdl[i - (NVARS + 1)] * Vl[ks * LDS_STRIDE + i];
                gbuf[ks] = g;
            }
            __syncthreads();

            if (part == 0 && ks < KDIM) {
                float ss = 0.f;
#pragma unroll
                for (int k2 = 0; k2 < KDIM; ++k2) ss += gbuf[k2] * gbuf[k2];
                float inv = -1.f / (sqrtf(ss) + EPSF);
                Vl[ks * LDS_STRIDE + i] = gbuf[ks] * inv;
            }
            if (tid < 32) grow_wait();   // next row resident before barrier
            __syncthreads();
            cur ^= 1;
        }
    }

    // zo = arccos(clip(-V[:, :, 1:513]^T v0)) / pi
    for (int j = tid; j < NVARS; j += SWT) {
        float t = 0.f;
#pragma unroll
        for (int k = 0; k < KDIM; ++k)
            t = fmaf(Vl[k * LDS_STRIDE + (1 + j)], v0l[k], t);
        t = -t;
        t = fminf(fmaxf(t, -1.f + 1e-6f), 1.f - 1e-6f);
        out[b * NVARS + j] = acosf(t) * (1.f / PI_F);
    }
}

// ---------------------------------------------------------------- launch
extern "C" void kernel_launch(void* const* d_in, const int* in_sizes, int n_in,
                              void* d_out, int out_size, void* d_ws, size_t ws_size,
                              hipStream_t stream) {
    const float* x      = (const float*)d_in[0];   // (16,16,32)
    const float* S      = (const float*)d_in[1];   // (600,813)
    const float* v_rand = (const float*)d_in[2];   // (41,813)
    float* out = (float*)d_out;
    float* ws  = (float*)d_ws;

    float* Vr    = ws + OFF_VR;
    float* P     = ws + OFF_P;
    float* gdiag = ws + OFF_GDIAG;
    float* G     = ws + OFF_G;
    float* V     = ws + OFF_V;
    float* Spad  = ws + OFF_SPAD;   // aliases V; consumed before V is written

    satnet_pad_s<<<(MCLS * NPAD + 255) / 256, 256, 0, stream>>>(S, Spad);
    satnet_norm_cols<<<(NTOT + 255) / 256, 256, 0, stream>>>(v_rand, Vr);
    satnet_project<<<(NVARS + 255) / 256, 256, 0, stream>>>(Vr, P);

    int ntiles = GTILES * GTILES;
    satnet_gram_wmma<<<(ntiles + 7) / 8, 256, 0, stream>>>(Spad, G);
    satnet_gdiag<<<(NPAD + 255) / 256, 256, 0, stream>>>(G, gdiag);

    satnet_init_v<<<(BATCH * NTOT + 255) / 256, 256, 0, stream>>>(x, Vr, P, V);

    size_t smem = (size_t)(SM_VL + KDIM * LDS_STRIDE) * sizeof(float);
    satnet_sweep<<<BATCH, SWT, smem, stream>>>(Vr, G, gdiag, V, out);
}